// UnitCellTlista_78975858639051
// MI455X (gfx1250) — compile-verified
//
#include <hip/hip_runtime.h>
#include <math.h>

typedef float v2f __attribute__((ext_vector_type(2)));
typedef float v8f __attribute__((ext_vector_type(8)));

#define BETA_ST 0.01f
#define EPS_ST  1e-12f

// ---------------------------------------------------------------------------
// Kernel 1: materialize the complex Toeplitz matrix W1[i,j] = v[N-1 + i - j]
// into workspace (8 MB total for re+im; stays resident in the 192MB L2).
// ---------------------------------------------------------------------------
__global__ void toeplitz_fill(const float* __restrict__ vre,
                              const float* __restrict__ vim,
                              float* __restrict__ w1re,
                              float* __restrict__ w1im,
                              int N) {
    int idx = blockIdx.x * blockDim.x + threadIdx.x;
    int total = N * N;
    if (idx >= total) return;
    int i = idx / N;
    int j = idx - i * N;
    int s = (N - 1) + i - j;
    w1re[idx] = vre[s];
    w1im[idx] = vim[s];
}

// ---------------------------------------------------------------------------
// fp32 WMMA helper: D = A(16x4) * B(4x16) + C(16x16), f32 everywhere.
// ---------------------------------------------------------------------------
__device__ __forceinline__ v8f wmma_f32(v2f a, v2f b, v8f c) {
    return __builtin_amdgcn_wmma_f32_16x16x4_f32(false, a, false, b,
                                                 (short)0, c, false, false);
}

// ---------------------------------------------------------------------------
// Kernel 2: fused complex GEMM  Z = W1@x + W2@y  + complex soft-threshold.
//
// 3M (Gauss) complex product per WMMA tile:
//   P1 = Ar*Br, P2 = Ai*Bi, P3 = (Ar+Ai)*(Br+Bi)
//   re = P1 - P2 ; im = P3 - P1 - P2
// -> 12 WMMAs + 4 v_pk_add_f32 per k-step (vs 16 WMMAs for the 4M form).
// Matrix-pipe bound => ~25% fewer matrix ops.
//
// Wave32: each wave computes a 32x32 complex output tile (2x2 WMMA tiles,
// 3 f32 accumulator chains each = 96 acc VGPRs).
// Block: 8 waves arranged 4(row)x2(col) -> 128x64; grid (B/64, N/128).
// ---------------------------------------------------------------------------
__global__ __launch_bounds__(256)
void cplx_gemm_softthresh(const float* __restrict__ W1re,
                          const float* __restrict__ W1im,
                          const float* __restrict__ W2re,
                          const float* __restrict__ W2im,
                          const float* __restrict__ xre,
                          const float* __restrict__ xim,
                          const float* __restrict__ yre,
                          const float* __restrict__ yim,
                          float* __restrict__ out,
                          int N, int M, int B) {
    const int lane = threadIdx.x & 31;
    const int wave = threadIdx.x >> 5;
    const int wr   = wave >> 1;               // 0..3 row group
    const int wc   = wave & 1;                // 0..1 col group
    const int row0 = blockIdx.y * 128 + wr * 32;
    const int col0 = blockIdx.x * 64  + wc * 32;

    const int lh = lane >> 4;                 // half-wave -> K pair (0: K=0,1  1: K=2,3)
    const int lm = lane & 15;                 // row (A) / col (B) within 16

    v8f p1[2][2], p2[2][2], p3[2][2];
#pragma unroll
    for (int t = 0; t < 2; ++t)
#pragma unroll
        for (int u = 0; u < 2; ++u) {
            v8f z = {0.f, 0.f, 0.f, 0.f, 0.f, 0.f, 0.f, 0.f};
            p1[t][u] = z; p2[t][u] = z; p3[t][u] = z;
        }

    // ---------------- Phase A: Z += W1 @ x   (K = N) ----------------
    {
        const unsigned arow = (unsigned)((row0 + lm) * N + 2 * lh);
        const unsigned bcol = (unsigned)(col0 + lm + 2 * lh * B);

#pragma unroll 4
        for (int k0 = 0; k0 < N; k0 += 4) {
            const unsigned ao = arow + (unsigned)k0;
            const unsigned bo = bcol + (unsigned)(k0 * B);
            v2f Ar0 = *(const v2f*)(W1re + ao);
            v2f Ai0 = *(const v2f*)(W1im + ao);
            v2f Ar1 = *(const v2f*)(W1re + ao + 16u * N);
            v2f Ai1 = *(const v2f*)(W1im + ao + 16u * N);
            v2f Br0 = {xre[bo],       xre[bo + B]};
            v2f Bi0 = {xim[bo],       xim[bo + B]};
            v2f Br1 = {xre[bo + 16u], xre[bo + B + 16u]};
            v2f Bi1 = {xim[bo + 16u], xim[bo + B + 16u]};

            v2f As0 = Ar0 + Ai0;     // v_pk_add_f32
            v2f As1 = Ar1 + Ai1;
            v2f Bs0 = Br0 + Bi0;
            v2f Bs1 = Br1 + Bi1;

            p1[0][0] = wmma_f32(Ar0, Br0, p1[0][0]);
            p2[0][0] = wmma_f32(Ai0, Bi0, p2[0][0]);
            p3[0][0] = wmma_f32(As0, Bs0, p3[0][0]);
            p1[0][1] = wmma_f32(Ar0, Br1, p1[0][1]);
            p2[0][1] = wmma_f32(Ai0, Bi1, p2[0][1]);
            p3[0][1] = wmma_f32(As0, Bs1, p3[0][1]);
            p1[1][0] = wmma_f32(Ar1, Br0, p1[1][0]);
            p2[1][0] = wmma_f32(Ai1, Bi0, p2[1][0]);
            p3[1][0] = wmma_f32(As1, Bs0, p3[1][0]);
            p1[1][1] = wmma_f32(Ar1, Br1, p1[1][1]);
            p2[1][1] = wmma_f32(Ai1, Bi1, p2[1][1]);
            p3[1][1] = wmma_f32(As1, Bs1, p3[1][1]);
        }
    }

    // ---------------- Phase B: Z += W2 @ y   (K = M) ----------------
    {
        const unsigned arow = (unsigned)((row0 + lm) * M + 2 * lh);
        const unsigned bcol = (unsigned)(col0 + lm + 2 * lh * B);

#pragma unroll 4
        for (int k0 = 0; k0 < M; k0 += 4) {
            const unsigned ao = arow + (unsigned)k0;
            const unsigned bo = bcol + (unsigned)(k0 * B);
            v2f Ar0 = *(const v2f*)(W2re + ao);
            v2f Ai0 = *(const v2f*)(W2im + ao);
            v2f Ar1 = *(const v2f*)(W2re + ao + 16u * M);
            v2f Ai1 = *(const v2f*)(W2im + ao + 16u * M);
            v2f Br0 = {yre[bo],       yre[bo + B]};
            v2f Bi0 = {yim[bo],       yim[bo + B]};
            v2f Br1 = {yre[bo + 16u], yre[bo + B + 16u]};
            v2f Bi1 = {yim[bo + 16u], yim[bo + B + 16u]};

            v2f As0 = Ar0 + Ai0;
            v2f As1 = Ar1 + Ai1;
            v2f Bs0 = Br0 + Bi0;
            v2f Bs1 = Br1 + Bi1;

            p1[0][0] = wmma_f32(Ar0, Br0, p1[0][0]);
            p2[0][0] = wmma_f32(Ai0, Bi0, p2[0][0]);
            p3[0][0] = wmma_f32(As0, Bs0, p3[0][0]);
            p1[0][1] = wmma_f32(Ar0, Br1, p1[0][1]);
            p2[0][1] = wmma_f32(Ai0, Bi1, p2[0][1]);
            p3[0][1] = wmma_f32(As0, Bs1, p3[0][1]);
            p1[1][0] = wmma_f32(Ar1, Br0, p1[1][0]);
            p2[1][0] = wmma_f32(Ai1, Bi0, p2[1][0]);
            p3[1][0] = wmma_f32(As1, Bs0, p3[1][0]);
            p1[1][1] = wmma_f32(Ar1, Br1, p1[1][1]);
            p2[1][1] = wmma_f32(Ai1, Bi1, p2[1][1]);
            p3[1][1] = wmma_f32(As1, Bs1, p3[1][1]);
        }
    }

    // ------------- Epilogue: complex soft-threshold + store -------------
    // re = P1 - P2 ; im = P3 - P1 - P2
    // C/D layout: VGPR e -> row M = e + 8*lh (lanes 16-31 shift by 8), col N = lm.
#pragma unroll
    for (int t = 0; t < 2; ++t)
#pragma unroll
        for (int u = 0; u < 2; ++u) {
            v8f re = p1[t][u] - p2[t][u];
            v8f im = p3[t][u] - p1[t][u] - p2[t][u];
            const int ocol = col0 + u * 16 + lm;
#pragma unroll
            for (int e = 0; e < 8; ++e) {
                const int orow = row0 + t * 16 + e + 8 * lh;
                float r = re[e], i = im[e];
                float mag   = sqrtf(r * r + i * i);
                float scale = fmaxf(mag - BETA_ST, 0.0f) / fmaxf(mag, EPS_ST);
                float2 val  = make_float2(r * scale, i * scale);
                const unsigned oofs = (unsigned)(2 * (orow * B + ocol));
                *(float2*)(out + oofs) = val;
            }
        }
}

// ---------------------------------------------------------------------------
extern "C" void kernel_launch(void* const* d_in, const int* in_sizes, int n_in,
                              void* d_out, int out_size, void* d_ws, size_t ws_size,
                              hipStream_t stream) {
    const float* vre  = (const float*)d_in[0];
    const float* vim  = (const float*)d_in[1];
    const float* W2re = (const float*)d_in[2];
    const float* W2im = (const float*)d_in[3];
    const float* xre  = (const float*)d_in[4];
    const float* xim  = (const float*)d_in[5];
    const float* yre  = (const float*)d_in[6];
    const float* yim  = (const float*)d_in[7];
    float* out = (float*)d_out;

    const int N = (in_sizes[0] + 1) / 2;   // 1024
    const int M = in_sizes[2] / N;         // 256
    const int B = in_sizes[4] / N;         // 1024

    // Workspace: W1_re [N*N] followed by W1_im [N*N]  (8 MB total)
    float* W1re = (float*)d_ws;
    float* W1im = W1re + (size_t)N * N;

    const int total = N * N;
    toeplitz_fill<<<(total + 255) / 256, 256, 0, stream>>>(vre, vim, W1re, W1im, N);

    dim3 grid(B / 64, N / 128);
    cplx_gemm_softthresh<<<grid, 256, 0, stream>>>(W1re, W1im, W2re, W2im,
                                                   xre, xim, yre, yim,
                                                   out, N, M, B);
}